// LSTM_Sequence_Prediction_80453327389022
// MI455X (gfx1250) — compile-verified
//
#include <hip/hip_runtime.h>

// LSTM seq2seq on gfx1250 using V_WMMA_F32_16X16X4_F32 (full fp32 precision;
// workload is recurrence-latency bound, weights are L2-resident, so fp32 WMMA
// is the right operating point on MI455X).

typedef __attribute__((ext_vector_type(2))) float v2f;
typedef __attribute__((ext_vector_type(8))) float v8f;

#define SLEN   256
#define BATCH  64
#define FEAT   256
#define HID    768
#define NLAYER 3
#define TLEN   32
#define GATES  (4 * HID)        // 3072
#define BH     (BATCH * HID)    // 49152

__device__ __forceinline__ float sigf(float x) {
  return 1.0f / (1.0f + __expf(-x));
}

// One block = one (batch-tile bt, hidden-col-tile jt, layer z).
// 4 waves; wave g computes the 16x16 tile of gate g (rows n = g*HID + jt*16 ..)
// via f32 WMMA over K = Kx (x part) + Kh (h part). Gate tiles meet in LDS,
// then the block applies the LSTM cell elementwise and writes h (and c).
__global__ __launch_bounds__(128) void lstm_cell_wmma(
    const float* __restrict__ xin, int Kx,          // (BATCH, Kx) or null
    const float* __restrict__ hin, int Kh,          // (BATCH, Kh) [+ z*BATCH*Kh]
    const float* __restrict__ Wx,                   // (GATES, Kx)
    const float* __restrict__ Wh,                   // (GATES, Kh) [+ z*GATES*Kh]
    const float* __restrict__ b1,                   // (GATES) [+ z*GATES]
    const float* __restrict__ b2,                   // (GATES) [+ z*GATES]
    const float* __restrict__ c_in,                 // (BATCH, HID) or null -> 0
    float* __restrict__ h_out,                      // (BATCH, HID) [+ z*BH]
    float* __restrict__ c_out,                      // (BATCH, HID) or null
    float* __restrict__ h_out2)                     // optional second copy
{
  const int lane = threadIdx.x & 31;
  const int g    = threadIdx.x >> 5;   // gate index 0..3 (i,f,g,o)
  const int jt   = blockIdx.x;         // hidden-col tile 0..47
  const int bt   = blockIdx.y;         // batch tile 0..3
  const int z    = blockIdx.z;         // layer (decoder); 0 for encoder

  Wh    += (size_t)z * GATES * Kh;
  b1    += (size_t)z * GATES;
  b2    += (size_t)z * GATES;
  hin   += (size_t)z * BATCH * Kh;
  h_out += (size_t)z * BH;

  // A layout (16x4 f32): lane holds A[m = lane&15][k = (lane>>4)*2 + {0,1}]
  // B layout (4x16 f32): lane holds B[k = (lane>>4)*2 + {0,1}][n = lane&15]
  const int l15 = lane & 15;
  const int kk  = (lane >> 4) * 2;
  const int wrow = g * HID + jt * 16 + l15;  // weight/bias row for col n=l15

  v8f acc = {};

  if (xin) {
    const float* arow = xin + (size_t)(bt * 16 + l15) * Kx + kk;
    const float* brow = Wx  + (size_t)wrow * Kx + kk;
#pragma unroll 4
    for (int k0 = 0; k0 < Kx; k0 += 4) {
      v2f a = *(const v2f*)(arow + k0);
      v2f b = *(const v2f*)(brow + k0);
      acc = __builtin_amdgcn_wmma_f32_16x16x4_f32(false, a, false, b,
                                                  (short)0, acc, false, false);
    }
  }
  {
    const float* arow = hin + (size_t)(bt * 16 + l15) * Kh + kk;
    const float* brow = Wh  + (size_t)wrow * Kh + kk;
#pragma unroll 4
    for (int k0 = 0; k0 < Kh; k0 += 4) {
      v2f a = *(const v2f*)(arow + k0);
      v2f b = *(const v2f*)(brow + k0);
      acc = __builtin_amdgcn_wmma_f32_16x16x4_f32(false, a, false, b,
                                                  (short)0, acc, false, false);
    }
  }

  // C/D layout: lane holds D[m = r + 8*(lane>>4)][n = lane&15] in vgpr r.
  __shared__ float lds[4 * 256];
  const float bias = b1[wrow] + b2[wrow];
#pragma unroll
  for (int r = 0; r < 8; ++r) {
    int mrow = r + 8 * (lane >> 4);
    lds[g * 256 + mrow * 16 + l15] = acc[r] + bias;
  }
  __syncthreads();

  for (int e = threadIdx.x; e < 256; e += 128) {
    float iv = lds[        e];
    float fv = lds[256  + e];
    float gv = lds[512  + e];
    float ov = lds[768  + e];
    int mm = e >> 4, nn = e & 15;
    int b = bt * 16 + mm;
    int j = jt * 16 + nn;
    float cp = c_in ? c_in[(size_t)b * HID + j] : 0.0f;
    float cn = sigf(fv) * cp + sigf(iv) * tanhf(gv);
    float hv = sigf(ov) * tanhf(cn);
    if (c_out)  c_out [(size_t)b * HID + j] = cn;
    h_out[(size_t)b * HID + j] = hv;
    if (h_out2) h_out2[(size_t)b * HID + j] = hv;
  }
}

// out[b, f] = hin[b, :] . W[f, :] + bias[f]   (M=64, N=256, K=768)
__global__ __launch_bounds__(32) void head_wmma(
    const float* __restrict__ hin,   // (BATCH, HID)
    const float* __restrict__ W,     // (FEAT, HID)
    const float* __restrict__ bias,  // (FEAT)
    float* __restrict__ out)         // (BATCH, FEAT)
{
  const int lane = threadIdx.x & 31;
  const int ft = blockIdx.x;  // 0..15
  const int bt = blockIdx.y;  // 0..3
  const int l15 = lane & 15;
  const int kk  = (lane >> 4) * 2;

  const float* arow = hin + (size_t)(bt * 16 + l15) * HID + kk;
  const float* brow = W   + (size_t)(ft * 16 + l15) * HID + kk;
  v8f acc = {};
#pragma unroll 4
  for (int k0 = 0; k0 < HID; k0 += 4) {
    v2f a = *(const v2f*)(arow + k0);
    v2f b = *(const v2f*)(brow + k0);
    acc = __builtin_amdgcn_wmma_f32_16x16x4_f32(false, a, false, b,
                                                (short)0, acc, false, false);
  }
  const float bb = bias[ft * 16 + l15];
#pragma unroll
  for (int r = 0; r < 8; ++r) {
    int row = r + 8 * (lane >> 4);
    out[(size_t)(bt * 16 + row) * FEAT + ft * 16 + l15] = acc[r] + bb;
  }
}

extern "C" void kernel_launch(void* const* d_in, const int* in_sizes, int n_in,
                              void* d_out, int out_size, void* d_ws, size_t ws_size,
                              hipStream_t stream)
{
  const float* x     = (const float*)d_in[0];
  const float* eWih0 = (const float*)d_in[1];   // (4H, F)
  const float* eWhh0 = (const float*)d_in[2];   // (4H, H)
  const float* ebih0 = (const float*)d_in[3];
  const float* ebhh0 = (const float*)d_in[4];
  const float* eWih  = (const float*)d_in[5];   // (L-1, 4H, H)
  /* d_in[6] e_Whh: unused (zero initial hidden, seq-len-1 layers) */
  const float* ebih  = (const float*)d_in[7];
  const float* ebhh  = (const float*)d_in[8];
  const float* dWih  = (const float*)d_in[9];   // (L, 4H, H)
  /* d_in[10] d_Whh: unused (decoder bug: fresh zero state each step) */
  const float* dbih  = (const float*)d_in[11];
  const float* dbhh  = (const float*)d_in[12];
  const float* linW  = (const float*)d_in[13];  // (F, H)
  const float* linb  = (const float*)d_in[14];

  float* out     = (float*)d_out;
  float* enc_out = out + (size_t)TLEN * BATCH * FEAT;  // encoder_hidden (L,B,H)

  float* wsf    = (float*)d_ws;
  float* h_ping = wsf;                         // (B,H)
  float* cbuf   = wsf + BH;                    // (B,H)
  float* dec0   = wsf + 2 * BH;                // (L,B,H)
  float* h_pong = wsf + 2 * BH + NLAYER * BH;  // (B,H)
  float* dec1   = h_pong + BH;                 // (L,B,H)

  // zero h0, c0, decoder hs0 (h_ping, cbuf, dec0 are contiguous)
  hipMemsetAsync(wsf, 0, (size_t)(2 * BH + NLAYER * BH) * sizeof(float), stream);

  // --- encoder layer 0: 256 sequential steps, ping/pong h, c in place ---
  float* hb[2] = { h_ping, h_pong };
  for (int s = 0; s < SLEN; ++s) {
    lstm_cell_wmma<<<dim3(48, 4, 1), 128, 0, stream>>>(
        x + (size_t)s * BATCH * FEAT, FEAT,
        hb[s & 1], HID,
        eWih0, eWhh0, ebih0, ebhh0,
        cbuf, hb[(s + 1) & 1], cbuf, nullptr);
  }
  // final h is in hb[0] (SLEN even) -> encoder_hidden[0]
  hipMemcpyAsync(enc_out, h_ping, (size_t)BH * sizeof(float),
                 hipMemcpyDeviceToDevice, stream);

  // --- encoder layers 1..2: single cell, zero state ---
  lstm_cell_wmma<<<dim3(48, 4, 1), 128, 0, stream>>>(
      nullptr, 0, h_ping, HID, nullptr,
      eWih, ebih, ebhh,
      nullptr, h_pong, nullptr, enc_out + BH);
  lstm_cell_wmma<<<dim3(48, 4, 1), 128, 0, stream>>>(
      nullptr, 0, h_pong, HID, nullptr,
      eWih + (size_t)GATES * HID, ebih + GATES, ebhh + GATES,
      nullptr, cbuf /*reuse*/, nullptr, enc_out + 2 * BH);

  // --- decoder: T steps, 3 independent layers per step (grid z) + head ---
  float* db[2] = { dec0, dec1 };
  for (int t = 0; t < TLEN; ++t) {
    lstm_cell_wmma<<<dim3(48, 4, NLAYER), 128, 0, stream>>>(
        nullptr, 0, db[t & 1], HID, nullptr,
        dWih, dbih, dbhh,
        nullptr, db[(t + 1) & 1], nullptr, nullptr);
    head_wmma<<<dim3(16, 4, 1), 32, 0, stream>>>(
        db[(t + 1) & 1] + 2 * BH, linW, linb,
        out + (size_t)t * BATCH * FEAT);
  }

  (void)in_sizes; (void)n_in; (void)out_size; (void)ws_size;
}